// HbvStatic_53188874993969
// MI455X (gfx1250) — compile-verified
//
#include <hip/hip_runtime.h>
#include <cstdint>
#include <cstddef>

#define N_PHY   12
#define NMUL    4
#define WARM_UP 365
#define LENF    15
#define TOUT    16      // outputs per block along time in conv kernel
#define GTILE   256     // gauges per block in conv kernel
#define SCAN_BLOCK 32   // one wave per block in scan kernel
#define RING    16      // prefetch depth (rows in flight) in scan kernel

__constant__ float c_lb[N_PHY] = {1.0f, 50.0f, 0.05f, 0.01f, 0.001f, 0.2f,
                                  0.0f, 0.0f, -2.5f, 0.5f, 0.0f, 0.0f};
__constant__ float c_ub[N_PHY] = {6.0f, 1000.0f, 0.9f, 0.5f, 0.2f, 1.0f,
                                  10.0f, 100.0f, 2.5f, 10.0f, 0.1f, 0.2f};

// ---------------------------------------------------------------------------
// Kernel 1: HBV recurrence. One WAVE per block (no barriers); the wave keeps a
// RING-deep LDS ring buffer of forcing rows filled with CDNA5 async
// global->LDS loads (ASYNCcnt-tracked, in-order completion), so the serial
// recurrence never stalls on an HBM miss. 4 multipliers unrolled in registers
// for ILP on the dependent chain. Writes qs[t*G+g] = mean_m q.
// ---------------------------------------------------------------------------
__global__ void __launch_bounds__(SCAN_BLOCK)
hbv_scan_kernel(const float* __restrict__ xphy,
                const float* __restrict__ params,
                float* __restrict__ qs,
                int T, int G)
{
    __shared__ float ring[RING * SCAN_BLOCK * 3];   // 6 KB, written only by async DMA

    const int lane = threadIdx.x;
    const int g  = blockIdx.x * SCAN_BLOCK + lane;
    const int gl = g < G ? g : G - 1;               // clamped for all loads

    const float* pr = params + (size_t)gl * (N_PHY * NMUL + 2);

    float beta[NMUL], fc[NMUL], inv_fc[NMUL], k0[NMUL], k1[NMUL], k2[NMUL];
    float inv_lpfc[NMUL], perc[NMUL], uzl[NMUL], ttp[NMUL], cfmax[NMUL];
    float cfrcf[NMUL], cwh[NMUL];

#pragma unroll
    for (int m = 0; m < NMUL; ++m) {
        float r[N_PHY];
#pragma unroll
        for (int i = 0; i < N_PHY; ++i)
            r[i] = c_lb[i] + pr[i * NMUL + m] * (c_ub[i] - c_lb[i]);
        beta[m]  = r[0];
        fc[m]    = r[1];
        k0[m]    = r[2];
        k1[m]    = r[3];
        k2[m]    = r[4];
        float lp = r[5];
        perc[m]  = r[6];
        uzl[m]   = r[7];
        ttp[m]   = r[8];
        cfmax[m] = r[9];
        cfrcf[m] = r[10] * r[9];   // cfr * cfmax
        cwh[m]   = r[11];
        inv_fc[m]   = 1.0f / fc[m];
        inv_lpfc[m] = 1.0f / (lp * fc[m]);
    }

    float sp[NMUL], mw[NMUL], sm[NMUL], suz[NMUL], slz[NMUL];
#pragma unroll
    for (int m = 0; m < NMUL; ++m) { sp[m] = mw[m] = sm[m] = suz[m] = slz[m] = 0.001f; }

    const unsigned ldsLane = (unsigned)(uintptr_t)(&ring[lane * 3]);

    // Prologue: fill the ring (rows 0..RING-1): 3 async b32 per lane per row.
    // Steady-state outstanding ASYNCcnt = 3*RING = 48 (< 63 counter limit).
#pragma unroll
    for (int r = 0; r < RING; ++r) {
        const int tr = r < T ? r : T - 1;
        const unsigned la = ldsLane + (unsigned)(r * SCAN_BLOCK * 3 * 4);
        const unsigned long long ga =
            (unsigned long long)(uintptr_t)(xphy + ((size_t)tr * G + gl) * 3);
        asm volatile("global_load_async_to_lds_b32 %0, %1, off"          :: "v"(la), "v"(ga) : "memory");
        asm volatile("global_load_async_to_lds_b32 %0, %1, off offset:4" :: "v"(la), "v"(ga) : "memory");
        asm volatile("global_load_async_to_lds_b32 %0, %1, off offset:8" :: "v"(la), "v"(ga) : "memory");
    }

    for (int t = 0; t < T; ++t) {
        // Async loads complete in order: <= 3*(RING-1)=45 outstanding means
        // the 3 loads of row t have landed in LDS.
        asm volatile("s_wait_asynccnt 0x2D" ::: "memory");

        const int slot = t & (RING - 1);
        const float* l = &ring[slot * (SCAN_BLOCK * 3) + lane * 3];
        const float Pt   = l[0];
        const float Tt   = l[1];
        const float PETt = l[2];

        // Make sure our DS reads of this slot retired before the async engine
        // is allowed to overwrite it with row t+RING.
        asm volatile("s_wait_dscnt 0x0" ::: "memory");
        {
            int tr = t + RING; tr = tr < T ? tr : T - 1;
            const unsigned la = ldsLane + (unsigned)(slot * SCAN_BLOCK * 3 * 4);
            const unsigned long long ga =
                (unsigned long long)(uintptr_t)(xphy + ((size_t)tr * G + gl) * 3);
            asm volatile("global_load_async_to_lds_b32 %0, %1, off"          :: "v"(la), "v"(ga) : "memory");
            asm volatile("global_load_async_to_lds_b32 %0, %1, off offset:4" :: "v"(la), "v"(ga) : "memory");
            asm volatile("global_load_async_to_lds_b32 %0, %1, off offset:8" :: "v"(la), "v"(ga) : "memory");
        }

        float qsum = 0.0f;
#pragma unroll
        for (int m = 0; m < NMUL; ++m) {
            const float rain = (Tt >= ttp[m]) ? Pt : 0.0f;
            const float snow = Pt - rain;
            sp[m] += snow;
            const float melt = fminf(fmaxf(cfmax[m] * (Tt - ttp[m]), 0.0f), sp[m]);
            mw[m] += melt;
            sp[m] -= melt;
            const float refreeze = fminf(fmaxf(cfrcf[m] * (ttp[m] - Tt), 0.0f), mw[m]);
            sp[m] += refreeze;
            mw[m] -= refreeze;
            const float tosoil = fmaxf(mw[m] - cwh[m] * sp[m], 0.0f);
            mw[m] -= tosoil;

            const float ratio = sm[m] * inv_fc[m];           // > 0 always (sm >= 1e-5)
            const float sw    = fminf(__powf(ratio, beta[m]), 1.0f);
            const float recharge = (rain + tosoil) * sw;
            sm[m] += rain + tosoil - recharge;
            const float excess = fmaxf(sm[m] - fc[m], 0.0f);
            sm[m] -= excess;
            const float evap  = fminf(sm[m] * inv_lpfc[m], 1.0f);
            const float etact = fminf(sm[m], PETt * evap);
            sm[m] = fmaxf(sm[m] - etact, 1e-5f);

            suz[m] += recharge + excess;
            const float prc = fminf(suz[m], perc[m]);
            suz[m] -= prc;
            const float q0 = k0[m] * fmaxf(suz[m] - uzl[m], 0.0f);
            suz[m] -= q0;
            const float q1 = k1[m] * suz[m];
            suz[m] -= q1;
            slz[m] += prc;
            const float q2 = k2[m] * slz[m];
            slz[m] -= q2;
            qsum += q0 + q1 + q2;
        }
        if (g < G)
            qs[(size_t)t * G + g] = qsum * 0.25f;
    }
}

// ---------------------------------------------------------------------------
// Kernel 2: per-gauge normalized gamma routing weights, stored [LENF][G].
// ---------------------------------------------------------------------------
__global__ void gamma_weights_kernel(const float* __restrict__ params,
                                     float* __restrict__ wbuf, int G)
{
    const int g = blockIdx.x * blockDim.x + threadIdx.x;
    if (g >= G) return;
    const float* pr = params + (size_t)g * (N_PHY * NMUL + 2);
    const float a  = pr[N_PHY * NMUL + 0] * 2.9f;   // ROUTE_A range [0, 2.9]
    const float b  = pr[N_PHY * NMUL + 1] * 6.5f;   // ROUTE_B range [0, 6.5]
    const float aa = fmaxf(a, 0.0f) + 0.1f;
    const float th = fmaxf(b, 0.0f) + 0.5f;
    const float denom = expf(lgammaf(aa)) * powf(th, aa);

    float wv[LENF];
    float s = 0.0f;
#pragma unroll
    for (int i = 0; i < LENF; ++i) {
        const float tg = (float)i + 0.5f;
        wv[i] = powf(tg, aa - 1.0f) * expf(-tg / th) / denom;
        s += wv[i];
    }
    const float inv_s = 1.0f / s;
#pragma unroll
    for (int i = 0; i < LENF; ++i)
        wbuf[(size_t)i * G + g] = wv[i] * inv_s;
}

// ---------------------------------------------------------------------------
// Kernel 3: 15-tap causal convolution. Stages a 32x256 fp32 tile of qs into
// LDS with CDNA5 async global->LDS B128 loads, then each thread produces 16
// outputs for one gauge (full unroll lets LLVM CSE the 240 LDS reads to ~30).
// ---------------------------------------------------------------------------
__global__ void __launch_bounds__(GTILE)
route_conv_kernel(const float* __restrict__ qs,
                  const float* __restrict__ wbuf,
                  float* __restrict__ out,
                  int T, int G, int Tout)
{
    __shared__ float tile[32 * GTILE];   // rows = time, cols = gauges (32 KB)

    const int tid   = threadIdx.x;
    const int gbase = blockIdx.x * GTILE;
    const int t0    = WARM_UP + blockIdx.y * TOUT;
    const int tbase = t0 - (LENF - 1);   // first qs row needed

    // Async-stage 32 rows x 256 floats: 8 sweeps x (256 lanes x 16B) = 32 KB.
#pragma unroll
    for (int s = 0; s < 8; ++s) {
        const int f   = s * (GTILE * 4) + tid * 4;  // float index in tile
        const int row = f >> 8;                     // f / GTILE
        const int col = f & (GTILE - 1);
        int tsrc = tbase + row;
        tsrc = tsrc < 0 ? 0 : (tsrc > T - 1 ? T - 1 : tsrc);
        const unsigned long long ga =
            (unsigned long long)(uintptr_t)(qs + ((size_t)tsrc * G + gbase + col));
        const unsigned ldsa = (unsigned)(uintptr_t)(&tile[f]);
        asm volatile("global_load_async_to_lds_b128 %0, %1, off"
                     :: "v"(ldsa), "v"(ga) : "memory");
    }
    asm volatile("s_wait_asynccnt 0x0" ::: "memory");
    __syncthreads();

    const int g  = gbase + tid;
    const int gc = g < G ? g : G - 1;
    float w[LENF];
#pragma unroll
    for (int i = 0; i < LENF; ++i)
        w[i] = wbuf[(size_t)i * G + gc];

#pragma unroll
    for (int tt = 0; tt < TOUT; ++tt) {
        float acc = 0.0f;
#pragma unroll
        for (int i = 0; i < LENF; ++i)
            acc += w[i] * tile[(LENF - 1 + tt - i) * GTILE + tid];
        const int t_out = blockIdx.y * TOUT + tt;
        if (g < G && t_out < Tout)
            out[(size_t)t_out * G + g] = acc;
    }
}

// ---------------------------------------------------------------------------
extern "C" void kernel_launch(void* const* d_in, const int* in_sizes, int n_in,
                              void* d_out, int out_size, void* d_ws, size_t ws_size,
                              hipStream_t stream)
{
    (void)n_in; (void)ws_size; (void)out_size;
    const float* xphy   = (const float*)d_in[0];   // [T, G, 3]
    const float* params = (const float*)d_in[1];   // [1, G, 50]
    float* out = (float*)d_out;                    // [T-WARM_UP, G, 1]

    const int G = in_sizes[1] / (N_PHY * NMUL + 2);
    const int T = in_sizes[0] / (3 * G);
    const int Tout = T - WARM_UP;

    // Workspace layout: qs [T*G] | 4096-float pad (conv edge overreach) | w [LENF*G]
    float* qs   = (float*)d_ws;
    float* wbuf = qs + ((size_t)T * G + 4096);

    {   // HBV scan: one wave per block, spread across WGPs
        const int blocks = (G + SCAN_BLOCK - 1) / SCAN_BLOCK;
        hbv_scan_kernel<<<blocks, SCAN_BLOCK, 0, stream>>>(xphy, params, qs, T, G);
    }
    {   // gamma routing weights
        const int threads = 256;
        const int blocks  = (G + threads - 1) / threads;
        gamma_weights_kernel<<<blocks, threads, 0, stream>>>(params, wbuf, G);
    }
    {   // routing convolution
        dim3 grid((G + GTILE - 1) / GTILE, (Tout + TOUT - 1) / TOUT);
        route_conv_kernel<<<grid, GTILE, 0, stream>>>(qs, wbuf, out, T, G, Tout);
    }
}